// lya_spec_81295140978958
// MI455X (gfx1250) — compile-verified
//
#include <hip/hip_runtime.h>
#include <stdint.h>

#define DIM   768
#define PANEL 16
#define NPANEL (DIM / PANEL)
#define DELTA_T 0.01f

typedef __attribute__((ext_vector_type(2))) float v2f;
typedef __attribute__((ext_vector_type(8))) float v8f;
typedef __attribute__((ext_vector_type(4))) unsigned int u32x4;
typedef __attribute__((ext_vector_type(8))) int i32x8;
typedef __attribute__((ext_vector_type(4))) int i32x4;

// D = A(16x4 f32) * B(4x16 f32) + C(16x16 f32) on the CDNA5 matrix pipe.
__device__ __forceinline__ v8f wmma4(v2f a, v2f b, v8f c) {
#if defined(__HIP_DEVICE_COMPILE__)
    return __builtin_amdgcn_wmma_f32_16x16x4_f32(false, a, false, b, (short)0, c,
                                                 false, false);
#else
    return c;
#endif
}

// TDM descriptor for a 768x16 f32 tile (tensor row stride 768 elements),
// LDS layout row-major -> matches Ep[r*16+c].
__device__ __forceinline__ void tdm_desc(u32x4& g0, i32x8& g1,
                                         const void* gaddr, unsigned lds_off) {
    unsigned long long ga = (unsigned long long)(uintptr_t)gaddr;
    g0[0] = 1u;                                   // count = 1 valid descriptor
    g0[1] = lds_off;                              // lds_addr (bytes)
    g0[2] = (unsigned)ga;                         // global_addr[31:0]
    g0[3] = (unsigned)(ga >> 32) | (2u << 30);    // global_addr[56:32] | type=2
    g1[0] = (int)(2u << 16);                      // data_size = 4B; no mask/pad
    g1[1] = (int)(768u << 16);                    // tensor_dim0 = 768 (lo16 @bit48)
    g1[2] = (int)(768u << 16);                    // tensor_dim1 = 768 (lo16 @bit80)
    g1[3] = (int)(16u << 16);                     // tile_dim0 = 16
    g1[4] = (int)768;                             // tile_dim1 = 768, tile_dim2 = 0
    g1[5] = (int)768;                             // tensor_dim0_stride = 768
    g1[6] = 0;
    g1[7] = 0;                                    // tensor_dim1_stride unused (2D)
}

// ------------------------------------------------- coef = (E_prev^T A_orig)/beta
__global__ void k_panel_coef(const float* __restrict__ E, const float* __restrict__ eye,
                             const float* __restrict__ beta, float* __restrict__ coef,
                             int pc0) {
    __shared__ float red[8 * 256];
    const int j0   = blockIdx.x * PANEL;
    const int lane = threadIdx.x & 31;
    const int wave = threadIdx.x >> 5;
    const int hl   = lane >> 4;      // half-wave selects K pair
    const int mn   = lane & 15;      // A: M index, B: N index

    v8f acc = {};
    const int rbeg = wave * 96;
    for (int rb = rbeg; rb < rbeg + 96; rb += 4) {
        const int r0 = rb + 2 * hl;
        v2f a, b;
        a.x = E[(size_t)(r0    ) * DIM + j0 + mn];    // A[m][k]  = E[r][j0+m]
        a.y = E[(size_t)(r0 + 1) * DIM + j0 + mn];
        b.x = eye[(size_t)(r0    ) * DIM + pc0 + mn]; // B[k][n] = eye[r][pc0+n]
        b.y = eye[(size_t)(r0 + 1) * DIM + pc0 + mn];
        acc = wmma4(a, b, acc);
    }
#pragma unroll
    for (int v = 0; v < 8; ++v) red[wave * 256 + lane * 8 + v] = acc[v];
    __syncthreads();

    const int t = threadIdx.x;
    const int m = t >> 4, n = t & 15;
    const int ln = ((m & 8) ? 16 : 0) + n;   // lane holding C(m,n)
    const int vg = m & 7;                    // vgpr holding row m
    float s = 0.f;
#pragma unroll
    for (int w = 0; w < 8; ++w) s += red[w * 256 + ln * 8 + vg];
    coef[(size_t)(j0 + m) * PANEL + n] = s / beta[j0 + m];
}

// ---------------------------------------- E[:,panel] = eye[:,panel] - E_prev @ coef
__global__ void k_panel_update(float* __restrict__ E, const float* __restrict__ eye,
                               const float* __restrict__ coef, int pc0) {
    const int r0   = blockIdx.x * PANEL;
    const int lane = threadIdx.x & 31;
    const int hl   = lane >> 4;
    const int mn   = lane & 15;

    v8f acc = {};
    for (int jb = 0; jb < pc0; jb += 4) {
        const int j = jb + 2 * hl;
        // A[m][k] = E[r0+m][j], K pair contiguous -> one b64 load
        const v2f a = *(const v2f*)(E + (size_t)(r0 + mn) * DIM + j);
        v2f b;
        b.x = coef[(size_t)(j    ) * PANEL + mn];     // B[k][n] = coef[j][n]
        b.y = coef[(size_t)(j + 1) * PANEL + mn];
        acc = wmma4(a, b, acc);
    }
#pragma unroll
    for (int v = 0; v < 8; ++v) {
        const int r = r0 + v + 8 * hl;                // D row layout (8 VGPRs)
        const size_t g = (size_t)r * DIM + pc0 + mn;
        E[g] = eye[g] - acc[v];
    }
}

// ------------------------- within-panel sequential classical GS (16 columns)
// Panel tile staged into LDS via the Tensor Data Mover; written back the same
// way. All reductions fixed-order -> deterministic.
__global__ void k_panel_gs(float* __restrict__ E, const float* __restrict__ src,
                           const float* __restrict__ eye,
                           float* __restrict__ beta, int pc0) {
    __shared__ float Ep[DIM * PANEL];   // 48 KB of the 320 KB WGP LDS
    __shared__ float red[256];
    __shared__ float coef_l[PANEL];
    __shared__ float beta_l[PANEL];

    const int t = threadIdx.x;
    const unsigned lds_off = (unsigned)(uintptr_t)Ep;

    if (t < 32) {  // one wave drives the TDM
#if defined(__HIP_DEVICE_COMPILE__)
        u32x4 g0; i32x8 g1; i32x4 z4 = {0, 0, 0, 0}; i32x8 z8 = {0, 0, 0, 0, 0, 0, 0, 0};
        tdm_desc(g0, g1, src + pc0, lds_off);
        __builtin_amdgcn_tensor_load_to_lds(g0, g1, z4, z4, z8, 0);
        __builtin_amdgcn_s_wait_tensorcnt(0);
#endif
    }
    __syncthreads();

    const int j     = t & 15;   // column this thread accumulates dots for
    const int chunk = t >> 4;   // 16 chunks x 48 rows

    for (int lk = 0; lk < PANEL; ++lk) {
        // d_j = sum_r Ep[r][j] * eye[r][pc0+lk]   (a = original column)
        float p = 0.f;
        const int rbase = chunk * 48;
        for (int rr = 0; rr < 48; ++rr) {
            const int r = rbase + rr;
            p += Ep[r * PANEL + j] * eye[(size_t)r * DIM + pc0 + lk];
        }
        red[t] = p;
        __syncthreads();
        if (t < PANEL) {
            float s = 0.f;
            for (int cc = 0; cc < 16; ++cc) s += red[cc * PANEL + t];
            coef_l[t] = (t < lk) ? (s / beta_l[t]) : 0.f;
        }
        __syncthreads();

        // Ep[:,lk] -= sum_{jj<lk} coef_l[jj] * Ep[:,jj]; accumulate beta
        float bp = 0.f;
        for (int rr = 0; rr < 3; ++rr) {
            const int r = t * 3 + rr;
            float v = Ep[r * PANEL + lk];
            for (int jj = 0; jj < lk; ++jj) v -= coef_l[jj] * Ep[r * PANEL + jj];
            Ep[r * PANEL + lk] = v;
            bp += v * v;
        }
        red[t] = bp;
        __syncthreads();
        if (t == 0) {
            float s = 0.f;
            for (int i = 0; i < 256; ++i) s += red[i];
            beta_l[lk] = s;
        }
        __syncthreads();
    }

    if (t < 32) {  // TDM store of the finished panel
#if defined(__HIP_DEVICE_COMPILE__)
        u32x4 g0; i32x8 g1; i32x4 z4 = {0, 0, 0, 0}; i32x8 z8 = {0, 0, 0, 0, 0, 0, 0, 0};
        tdm_desc(g0, g1, E + pc0, lds_off);
        __builtin_amdgcn_tensor_store_from_lds(g0, g1, z4, z4, z8, 0);
        __builtin_amdgcn_s_wait_tensorcnt(0);
#endif
    }
    if (t < PANEL) beta[pc0 + t] = beta_l[t];
}

// ------------------------------------------------------------------- epilogue
__global__ void k_finalize(const float* __restrict__ curr_t, const float* __restrict__ LE,
                           const float* __restrict__ beta, float* __restrict__ out) {
    const int k = blockIdx.x * blockDim.x + threadIdx.x;
    if (k >= DIM) return;
    const float tm    = curr_t[0];             // T_MARK == 0
    const float denom = tm + DELTA_T;
    const float logn  = 0.5f * logf(beta[k]);  // log(sqrt(beta))
    out[k] = (tm / denom) * LE[k] + (DELTA_T / denom) * logn;
}

extern "C" void kernel_launch(void* const* d_in, const int* in_sizes, int n_in,
                              void* d_out, int out_size, void* d_ws, size_t ws_size,
                              hipStream_t stream) {
    const float* curr_t = (const float*)d_in[0];
    const float* eye    = (const float*)d_in[1];
    const float* LE     = (const float*)d_in[2];
    float* out = (float*)d_out;

    float* E    = (float*)d_ws;          // 768*768 working matrix
    float* beta = E + DIM * DIM;         // 768 squared norms
    float* coef = beta + DIM;            // up to 768*16 projection coefs

    for (int p = 0; p < NPANEL; ++p) {
        const int pc0 = p * PANEL;
        if (p > 0) {
            k_panel_coef<<<p, 256, 0, stream>>>(E, eye, beta, coef, pc0);
            k_panel_update<<<NPANEL, 32, 0, stream>>>(E, eye, coef, pc0);
        }
        // panel 0 sources directly from eye; later panels from the projected E
        const float* src = (p == 0) ? eye : E;
        k_panel_gs<<<1, 256, 0, stream>>>(E, src, eye, beta, pc0);
    }

    k_finalize<<<(DIM + 255) / 256, 256, 0, stream>>>(curr_t, LE, beta, out);
}